// UDCTModule_71562745086439
// MI455X (gfx1250) — compile-verified
//
#include <hip/hip_runtime.h>

typedef float v2f __attribute__((ext_vector_type(2)));
typedef float v8f __attribute__((ext_vector_type(8)));

#define TWO_PI 6.28318530717958647692f

// Complex 32x32 matmul C = A * B using v_wmma_f32_16x16x4_f32 tiles.
// A, B, C are 32x32 row-major planar (re/im) arrays in LDS.
// 4 waves (128 threads): wave w computes output tile (w&1, w>>1).
// Complex combine: Cre += Are*Bre + (-Aim)*Bim ; Cim += Are*Bim + Aim*Bre
// (f32 WMMA has no A/B neg modifier, so the Aim fragment is negated in VALU).
__device__ __forceinline__ void cmatmul32(const float* __restrict__ Are,
                                          const float* __restrict__ Aim,
                                          const float* __restrict__ Bre,
                                          const float* __restrict__ Bim,
                                          float* __restrict__ Cre_out,
                                          float* __restrict__ Cim_out,
                                          int tid)
{
    const int lane  = tid & 31;
    const int wave  = tid >> 5;          // 0..3
    const int tm    = (wave & 1) * 16;   // tile row base (M)
    const int tn    = (wave >> 1) * 16;  // tile col base (N)
    const int r     = lane & 15;
    const int khalf = lane >> 4;         // 0: K={0,1}, 1: K={2,3}

    v8f cre = {};
    v8f cim = {};
    #pragma unroll
    for (int kk = 0; kk < 32; kk += 4) {
        const int ka   = kk + khalf * 2;
        const int arow = tm + r;
        const int bcol = tn + r;
        v2f are, aim, bre, bim;
        // A fragment: 16x4, lane holds A[row][ka], A[row][ka+1]
        are[0] = Are[arow * 32 + ka];     are[1] = Are[arow * 32 + ka + 1];
        aim[0] = Aim[arow * 32 + ka];     aim[1] = Aim[arow * 32 + ka + 1];
        // B fragment: 4x16, lane holds B[ka][col], B[ka+1][col]
        bre[0] = Bre[ka * 32 + bcol];     bre[1] = Bre[(ka + 1) * 32 + bcol];
        bim[0] = Bim[ka * 32 + bcol];     bim[1] = Bim[(ka + 1) * 32 + bcol];
        const v2f naim = -aim;            // VALU negate (no neg_a on f32 WMMA)

        cre = __builtin_amdgcn_wmma_f32_16x16x4_f32(false, are,  false, bre, (short)0, cre, false, false);
        cre = __builtin_amdgcn_wmma_f32_16x16x4_f32(false, naim, false, bim, (short)0, cre, false, false);
        cim = __builtin_amdgcn_wmma_f32_16x16x4_f32(false, are,  false, bim, (short)0, cim, false, false);
        cim = __builtin_amdgcn_wmma_f32_16x16x4_f32(false, aim,  false, bre, (short)0, cim, false, false);
    }
    // C/D fragment layout: VGPR v, lanes 0-15 -> M=v, lanes 16-31 -> M=v+8, N=lane&15
    #pragma unroll
    for (int v = 0; v < 8; ++v) {
        const int row = tm + v + 8 * khalf;
        const int col = tn + r;
        Cre_out[row * 32 + col] = cre[v];
        Cim_out[row * 32 + col] = cim[v];
    }
}

// One 1024-point FFT per block along an arbitrary-stride line.
// Four-step: view x[n], n = 32*n1 + n2 as M[n1][n2];
//   Y = DFT32 x M ; Y[k1][n2] *= W_1024^(sign*k1*n2) ; Z = Y x DFT32
//   X[k1 + 32*k2] = Z[k1][k2]
// iIm == nullptr  -> real input
// win != nullptr  -> multiply input by window (same indexing as input)
// oIm == nullptr  -> store real part only (non-temporal stream)
__global__ __launch_bounds__(128) void udct_fft1024_kernel(
    const float* __restrict__ iRe, const float* __restrict__ iIm,
    const float* __restrict__ win,
    float* __restrict__ oRe, float* __restrict__ oIm,
    long long iLineStride, long long iElemStride,
    long long oLineStride, long long oElemStride,
    float sign, float scale)
{
    __shared__ float sDre[1024], sDim[1024];   // DFT-32 matrix
    __shared__ float sXre[1024], sXim[1024];   // M, then Z
    __shared__ float sYre[1024], sYim[1024];   // Y

    const int tid = threadIdx.x;
    const long long ib = (long long)blockIdx.x * iLineStride;
    const long long ob = (long long)blockIdx.x * oLineStride;

    // Build DFT-32 matrix: D[j][k] = exp(sign * 2*pi*i * j*k / 32)
    const float w32 = sign * (TWO_PI / 32.0f);
    for (int idx = tid; idx < 1024; idx += 128) {
        const int j = idx >> 5, k = idx & 31;
        float s, c;
        __sincosf(w32 * (float)(j * k), &s, &c);
        sDre[idx] = c;
        sDim[idx] = s;
    }

    // Load line (optionally fused frequency-window multiply, NT streamed)
    for (int idx = tid; idx < 1024; idx += 128) {
        const long long g = ib + (long long)idx * iElemStride;
        float re = iRe[g];
        float im = iIm ? iIm[g] : 0.0f;
        if (win) {
            const float w = __builtin_nontemporal_load(&win[g]);
            re *= w;
            im *= w;
        }
        sXre[idx] = re;   // M[n1][n2] at idx = 32*n1 + n2
        sXim[idx] = im;
    }
    __syncthreads();

    // Stage 1: Y = D x M
    cmatmul32(sDre, sDim, sXre, sXim, sYre, sYim, tid);
    __syncthreads();

    // Twiddle: Y[k1][n2] *= exp(sign * 2*pi*i * k1*n2 / 1024)
    const float wN = sign * (TWO_PI / 1024.0f);
    for (int idx = tid; idx < 1024; idx += 128) {
        const int k1 = idx >> 5, n2 = idx & 31;
        float s, c;
        __sincosf(wN * (float)(k1 * n2), &s, &c);
        const float yr = sYre[idx], yi = sYim[idx];
        sYre[idx] = yr * c - yi * s;
        sYim[idx] = yr * s + yi * c;
    }
    __syncthreads();

    // Stage 2: Z = Y x D   (D symmetric, so this is the n2-DFT)
    cmatmul32(sYre, sYim, sDre, sDim, sXre, sXim, tid);
    __syncthreads();

    // Digit-reversed store: X[k1 + 32*k2] = Z[k1][k2]
    for (int f = tid; f < 1024; f += 128) {
        const int k1 = f & 31, k2 = f >> 5;
        const float zr = sXre[k1 * 32 + k2] * scale;
        const long long g = ob + (long long)f * oElemStride;
        if (oIm) {
            oRe[g] = zr;
            oIm[g] = sXim[k1 * 32 + k2] * scale;
        } else {
            __builtin_nontemporal_store(zr, &oRe[g]);
        }
    }
}

extern "C" void kernel_launch(void* const* d_in, const int* in_sizes, int n_in,
                              void* d_out, int out_size, void* d_ws, size_t ws_size,
                              hipStream_t stream) {
    const float* image   = (const float*)d_in[0];   // [1024,1024] f32
    const float* windows = (const float*)d_in[1];   // [43,1024,1024] f32
    float* out = (float*)d_out;                     // 43*1024*1024 f32

    const long long HW = 1024LL * 1024LL;
    const int nbands = (int)(out_size / HW);        // 43

    // Workspace: two complex planar 1024x1024 buffers (16 MB)
    float* aRe = (float*)d_ws;
    float* aIm = aRe + HW;
    float* bRe = aIm + HW;
    float* bIm = bRe + HW;

    dim3 grid(1024), block(128);

    // Pass 1: forward FFT along rows (real input) -> bufA
    udct_fft1024_kernel<<<grid, block, 0, stream>>>(
        image, nullptr, nullptr, aRe, aIm,
        1024, 1, 1024, 1, -1.0f, 1.0f);

    // Pass 2: forward FFT along columns -> bufB = F (L2-resident, 8 MB)
    udct_fft1024_kernel<<<grid, block, 0, stream>>>(
        aRe, aIm, nullptr, bRe, bIm,
        1, 1024, 1, 1024, -1.0f, 1.0f);

    for (int b = 0; b < nbands; ++b) {
        const float* w = windows + (long long)b * HW;
        // Pass 3: window multiply fused + inverse FFT along columns -> bufA
        udct_fft1024_kernel<<<grid, block, 0, stream>>>(
            bRe, bIm, w, aRe, aIm,
            1, 1024, 1, 1024, 1.0f, 1.0f / 1024.0f);
        // Pass 4: inverse FFT along rows, real part only -> out[b]
        udct_fft1024_kernel<<<grid, block, 0, stream>>>(
            aRe, aIm, nullptr, out + (long long)b * HW, nullptr,
            1024, 1, 1024, 1, 1.0f, 1.0f / 1024.0f);
    }
}